// Message_2628519985360
// MI455X (gfx1250) — compile-verified
//
#include <hip/hip_runtime.h>

typedef __attribute__((ext_vector_type(2))) float v2f;
typedef __attribute__((ext_vector_type(8))) float v8f;

#define FDIM 128
#define F3   384
#define NRBF 20
#define PI_F 3.14159265358979323846f
#define PI_OVER_CUT 0.62831853071795864769f  // pi / 5.0
#define CUT 5.0f

__device__ __forceinline__ v8f wmma_f32_k4(v2f a, v2f b, v8f c) {
  // D = A(16x4 f32) * B(4x16 f32) + C(16x16 f32)
  return __builtin_amdgcn_wmma_f32_16x16x4_f32(
      /*neg_a=*/false, a, /*neg_b=*/false, b,
      /*c_mod=*/(short)0, c, /*reuse_a=*/false, /*reuse_b=*/false);
}

__device__ __forceinline__ float fast_silu(float x) {
  // x * sigmoid(x); v_exp_f32 + v_rcp_f32 (single TRANS ops, no IEEE div chain)
  return x * __builtin_amdgcn_rcpf(1.0f + __expf(-x));
}

// Per-node fused transform:
//   t    = silu(node_s @ W1 + b1)              (16 x 128 per block)
//   h    = t @ W2 + b2                          (16 x 384)
//   filt = rbf(edge_dis) @ Wf + bf              (16 x 384)
//   g    = filt * cc * h                        (written to workspace)
__global__ void __launch_bounds__(256)
node_transform_kernel(const float* __restrict__ node_s,
                      const float* __restrict__ edge_dis,
                      const float* __restrict__ W1, const float* __restrict__ b1,
                      const float* __restrict__ W2, const float* __restrict__ b2,
                      const float* __restrict__ Wf, const float* __restrict__ bf,
                      float* __restrict__ g) {
  __shared__ float sA[16 * FDIM];   // node_s tile
  __shared__ float sT[16 * FDIM];   // silu(A@W1+b1) tile
  __shared__ float sR[16 * NRBF];   // rbf tile
  __shared__ float sCC[16];         // cosine cutoff per row

  const int tid  = threadIdx.x;
  const int wave = tid >> 5;
  const int lane = tid & 31;
  const int rowBase = blockIdx.x * 16;

  // Stage node_s tile (rows are contiguous -> straight float4 copy)
  {
    const float4* s4 = (const float4*)(node_s + (long)rowBase * FDIM);
    float4* d4 = (float4*)sA;
    for (int i = tid; i < 16 * FDIM / 4; i += 256) d4[i] = s4[i];
  }
  if (tid < 16) {
    float d = edge_dis[rowBase + tid];
    sCC[tid] = (d <= CUT) ? 0.5f * (__cosf(PI_F * d / CUT) + 1.0f) : 0.0f;
  }
  for (int t = tid; t < 16 * NRBF; t += 256) {
    int r = t / NRBF, k = t % NRBF;
    float d = edge_dis[rowBase + r];
    sR[t] = __sinf((float)(k + 1) * PI_OVER_CUT * d) * __builtin_amdgcn_rcpf(d);
  }
  __syncthreads();

  // WMMA fragment coordinates (ISA 7.12.2 layouts, wave32):
  //   A 16x4 f32: lanes 0-15 -> M=lane, K=k0+{0,1}; lanes 16-31 -> M=lane-16, K=k0+{2,3}
  //   B 4x16 f32: lanes 0-15 -> N=lane, K=k0+{0,1}; lanes 16-31 -> N=lane-16, K=k0+{2,3}
  //   C 16x16:    VGPR i -> M = i + (lane>=16 ? 8 : 0), N = lane&15
  const int mrow  = lane & 15;
  const int khalf = (lane >> 4) * 2;

  // ---- Stage 1: each of 8 waves computes one 16-col tile of t ----
  {
    const int c = wave * 16 + mrow;
    v8f acc = {};
    #pragma unroll 4
    for (int k0 = 0; k0 < FDIM; k0 += 4) {
      const int kb = k0 + khalf;
      v2f a; a.x = sA[mrow * FDIM + kb];  a.y = sA[mrow * FDIM + kb + 1];
      v2f b; b.x = W1[kb * FDIM + c];     b.y = W1[(kb + 1) * FDIM + c];
      acc = wmma_f32_k4(a, b, acc);
    }
    const float bias = b1[c];
    #pragma unroll
    for (int i = 0; i < 8; ++i) {
      const int r = i + (lane >> 4) * 8;
      sT[r * FDIM + c] = fast_silu(acc[i] + bias);
    }
  }
  __syncthreads();

  // ---- Stage 2: 24 col-tiles of h/filt; 3 tiles per wave ----
  for (int tno = 0; tno < 3; ++tno) {
    const int c = (wave * 3 + tno) * 16 + mrow;
    v8f acc = {};
    #pragma unroll 4
    for (int k0 = 0; k0 < FDIM; k0 += 4) {
      const int kb = k0 + khalf;
      v2f a; a.x = sT[mrow * FDIM + kb];  a.y = sT[mrow * FDIM + kb + 1];
      v2f b; b.x = W2[kb * F3 + c];       b.y = W2[(kb + 1) * F3 + c];
      acc = wmma_f32_k4(a, b, acc);
    }
    v8f facc = {};
    #pragma unroll
    for (int k0 = 0; k0 < NRBF; k0 += 4) {
      const int kb = k0 + khalf;
      v2f a; a.x = sR[mrow * NRBF + kb];  a.y = sR[mrow * NRBF + kb + 1];
      v2f b; b.x = Wf[kb * F3 + c];       b.y = Wf[(kb + 1) * F3 + c];
      facc = wmma_f32_k4(a, b, facc);
    }
    const float bias2 = b2[c], biasf = bf[c];
    #pragma unroll
    for (int i = 0; i < 8; ++i) {
      const int r = i + (lane >> 4) * 8;
      float h  = acc[i] + bias2;
      float fl = facc[i] + biasf;
      g[(long)(rowBase + r) * F3 + c] = fl * sCC[r] * h;
    }
  }
}

// Initialize outputs with the residual terms (out = node_vec / node_s copies);
// edge kernel then accumulates messages on top via atomics.
__global__ void init_out_kernel(const float4* __restrict__ vec4,
                                const float4* __restrict__ s4,
                                float4* __restrict__ out4,
                                long nVec4, long nS4) {
  const long stride = (long)gridDim.x * blockDim.x;
  for (long i = (long)blockIdx.x * blockDim.x + threadIdx.x; i < nVec4; i += stride)
    out4[i] = vec4[i];
  for (long i = (long)blockIdx.x * blockDim.x + threadIdx.x; i < nS4; i += stride)
    out4[nVec4 + i] = s4[i];
}

__device__ __forceinline__ void fadd_noret(float* p, float v) {
  (void)__hip_atomic_fetch_add(p, v, __ATOMIC_RELAXED, __HIP_MEMORY_SCOPE_AGENT);
}

// One wave per edge; lane handles 4 feature columns (c = lane*4 .. lane*4+3).
__global__ void __launch_bounds__(256)
edge_scatter_kernel(const long long* __restrict__ edge,
                    const float* __restrict__ g,
                    const float* __restrict__ node_vec,
                    float* __restrict__ out_vec,
                    float* __restrict__ out_s,
                    int E) {
  const int e = blockIdx.x * 8 + (threadIdx.x >> 5);
  if (e >= E) return;
  const int lane = threadIdx.x & 31;
  const long src = (long)edge[2 * e + 1];
  const long dst = (long)edge[2 * e + 0];

  const float4 gate = *(const float4*)(g + src * F3 + lane * 4);         // cols 0..127
  const float4 msg  = *(const float4*)(g + src * F3 + 256 + lane * 4);   // cols 256..383

  const float4* nv = (const float4*)(node_vec + src * F3 + lane * 12);   // (128,3) layout
  const float4 v0 = nv[0], v1 = nv[1], v2 = nv[2];

  float m[12];
  m[0]  = v0.x * gate.x; m[1]  = v0.y * gate.x; m[2]  = v0.z * gate.x;
  m[3]  = v0.w * gate.y; m[4]  = v1.x * gate.y; m[5]  = v1.y * gate.y;
  m[6]  = v1.z * gate.z; m[7]  = v1.w * gate.z; m[8]  = v2.x * gate.z;
  m[9]  = v2.y * gate.w; m[10] = v2.z * gate.w; m[11] = v2.w * gate.w;

  float* pv = out_vec + dst * F3 + lane * 12;
  #pragma unroll
  for (int j = 0; j < 12; ++j) fadd_noret(pv + j, m[j]);

  float* ps = out_s + dst * FDIM + lane * 4;
  fadd_noret(ps + 0, msg.x);
  fadd_noret(ps + 1, msg.y);
  fadd_noret(ps + 2, msg.z);
  fadd_noret(ps + 3, msg.w);
}

extern "C" void kernel_launch(void* const* d_in, const int* in_sizes, int n_in,
                              void* d_out, int out_size, void* d_ws, size_t ws_size,
                              hipStream_t stream) {
  const float*     node_s   = (const float*)d_in[0];
  const float*     node_vec = (const float*)d_in[1];
  const long long* edge     = (const long long*)d_in[2];
  // d_in[3] = edge_difference: unused by the reference
  const float*     edge_dis = (const float*)d_in[4];
  const float*     W1 = (const float*)d_in[5];
  const float*     b1 = (const float*)d_in[6];
  const float*     W2 = (const float*)d_in[7];
  const float*     b2 = (const float*)d_in[8];
  const float*     Wf = (const float*)d_in[9];
  const float*     bf = (const float*)d_in[10];

  const int N = in_sizes[0] / FDIM;   // 50000
  const int E = in_sizes[2] / 2;      // 800000

  float* g       = (float*)d_ws;                  // N*384 f32 = 76.8 MB scratch
  float* out_vec = (float*)d_out;                 // (N,128,3)
  float* out_s   = out_vec + (long)N * F3;        // (N,128)

  // 1) per-node fused GEMM/filter table (WMMA)
  node_transform_kernel<<<(N + 15) / 16, 256, 0, stream>>>(
      node_s, edge_dis, W1, b1, W2, b2, Wf, bf, g);

  // 2) residual init of outputs
  const long nVec4 = (long)N * F3 / 4;
  const long nS4   = (long)N * FDIM / 4;
  init_out_kernel<<<2048, 256, 0, stream>>>(
      (const float4*)node_vec, (const float4*)node_s, (float4*)d_out, nVec4, nS4);

  // 3) edge gather / scatter-add (atomic, memory-bound phase)
  edge_scatter_kernel<<<(E + 7) / 8, 256, 0, stream>>>(
      edge, g, node_vec, out_vec, out_s, E);
}